// IM_VIT_9517647528612
// MI455X (gfx1250) — compile-verified
//
#include <hip/hip_runtime.h>
#include <hip/hip_bf16.h>
#include <math.h>
#include <stdint.h>

typedef float v2f __attribute__((ext_vector_type(2)));
typedef float v8f __attribute__((ext_vector_type(8)));

#define NPAD 208   // 197 tokens padded to 13*16
#define NTOK 197
#define HD   48
#define NH   8
#define CDIM 384
#define BSZ  64
#define MROWS (BSZ*NTOK)   // 12608 = 16*788

// ---------------------------------------------------------------------------
// CDNA5 async global->LDS copy (bypasses VGPRs, tracked by ASYNCcnt).
// ldsPtr: pointer into __shared__ (low 32 bits of flat shared address are the
// LDS byte offset per the CDNA5 aperture mapping). 16B per lane.
// ---------------------------------------------------------------------------
__device__ __forceinline__ void async_copy_b128(void* ldsPtr, const float* g)
{
    uint32_t loff = (uint32_t)(uintptr_t)ldsPtr;
    asm volatile("global_load_async_to_lds_b128 %0, %1, off"
                 :: "v"(loff), "v"(g)
                 : "memory");
}
__device__ __forceinline__ void wait_async_le4()
{
    asm volatile("s_wait_asynccnt 0x4" ::: "memory");
}
__device__ __forceinline__ void wait_async_0()
{
    asm volatile("s_wait_asynccnt 0x0" ::: "memory");
}

// ---------------------------------------------------------------------------
// f32 WMMA GEMM: Out[M][N] = A[M][K] @ W[N][K]^T (+ bias)
// block = 256 threads (8 waves); block tile 128(M) x 128(N); K % 16 == 0,
// N % 128 == 0. Double-buffered LDS filled with async global->LDS b128 copies
// (4 async ops per wave per chunk -> s_wait_asynccnt 4 drains chunk c while
// chunk c+1 is still in flight).
// ---------------------------------------------------------------------------
__global__ __launch_bounds__(256) void gemm_wmma_f32(
    const float* __restrict__ A, const float* __restrict__ W,
    const float* __restrict__ bias, float* __restrict__ Out,
    int M, int N, int K)
{
    __shared__ float lsA[2][128 * 16];
    __shared__ float lsW[2][128 * 16];
    const int tid  = threadIdx.x;
    const int wave = tid >> 5;
    const int lane = tid & 31;
    const int lm   = lane & 15;
    const int hi   = lane >> 4;
    const int koff = hi * 2;
    const int m0   = blockIdx.y * 128;
    const int n0   = blockIdx.x * 128;
    const int nchunks = K >> 4;

    v8f acc[8] = {};

    // issue chunk 0 into buffer 0
    {
        #pragma unroll
        for (int i = 0; i < 2; ++i) {
            int idx4 = (tid + i * 256) * 4;        // float index in 128x16 tile
            int r = idx4 >> 4, cc = idx4 & 15;
            int gr = m0 + r; if (gr >= M) gr = M - 1;
            async_copy_b128(&lsA[0][idx4], &A[(size_t)gr * K + cc]);
            async_copy_b128(&lsW[0][idx4], &W[(size_t)(n0 + r) * K + cc]);
        }
    }

    for (int c = 0; c < nchunks; ++c) {
        const int cur = c & 1;
        if (c + 1 < nchunks) {
            const int k1 = (c + 1) << 4;
            #pragma unroll
            for (int i = 0; i < 2; ++i) {
                int idx4 = (tid + i * 256) * 4;
                int r = idx4 >> 4, cc = idx4 & 15;
                int gr = m0 + r; if (gr >= M) gr = M - 1;
                async_copy_b128(&lsA[cur ^ 1][idx4], &A[(size_t)gr * K + k1 + cc]);
                async_copy_b128(&lsW[cur ^ 1][idx4], &W[(size_t)(n0 + r) * K + k1 + cc]);
            }
            wait_async_le4();   // oldest 4 (chunk c) complete; chunk c+1 in flight
        } else {
            wait_async_0();     // last chunk: drain everything
        }
        __syncthreads();        // all waves' chunk-c data now visible in LDS

        #pragma unroll
        for (int kk = 0; kk < 16; kk += 4) {
            v2f a;
            a.x = lsA[cur][(wave * 16 + lm) * 16 + kk + koff];
            a.y = lsA[cur][(wave * 16 + lm) * 16 + kk + koff + 1];
            #pragma unroll
            for (int j = 0; j < 8; ++j) {
                v2f b;
                b.x = lsW[cur][(j * 16 + lm) * 16 + kk + koff];
                b.y = lsW[cur][(j * 16 + lm) * 16 + kk + koff + 1];
                acc[j] = __builtin_amdgcn_wmma_f32_16x16x4_f32(
                    false, a, false, b, (short)0, acc[j], false, false);
            }
        }
        __syncthreads();        // done reading buf[cur] before it is re-filled
    }

    #pragma unroll
    for (int j = 0; j < 8; ++j) {
        #pragma unroll
        for (int i = 0; i < 8; ++i) {
            int r = m0 + wave * 16 + i + (hi << 3);
            int cc = n0 + j * 16 + lm;
            if (r < M) {
                float v = acc[j][i];
                if (bias) v += bias[cc];
                Out[(size_t)r * N + cc] = v;
            }
        }
    }
}

// ---------------------------------------------------------------------------
// DynamicPosBias MLP: biases[841][2] -> pos[841][8]
// ---------------------------------------------------------------------------
__device__ inline void ln24_relu(float* t, const float* __restrict__ g,
                                 const float* __restrict__ b)
{
    float mu = 0.f;
    #pragma unroll
    for (int i = 0; i < 24; ++i) mu += t[i];
    mu *= (1.f / 24.f);
    float var = 0.f;
    #pragma unroll
    for (int i = 0; i < 24; ++i) { float d = t[i] - mu; var += d * d; }
    var *= (1.f / 24.f);
    float inv = rsqrtf(var + 1e-5f);
    #pragma unroll
    for (int i = 0; i < 24; ++i) {
        float y = (t[i] - mu) * inv * g[i] + b[i];
        t[i] = fmaxf(y, 0.f);
    }
}

__global__ __launch_bounds__(128) void pos_mlp(
    const float* __restrict__ biases,
    const float* __restrict__ ppw, const float* __restrict__ ppb,
    const float* __restrict__ g1, const float* __restrict__ b1,
    const float* __restrict__ w1, const float* __restrict__ bb1,
    const float* __restrict__ g2, const float* __restrict__ b2,
    const float* __restrict__ w2, const float* __restrict__ bb2,
    const float* __restrict__ g3, const float* __restrict__ b3,
    const float* __restrict__ w3, const float* __restrict__ bb3,
    float* __restrict__ pos)
{
    int r = blockIdx.x * 128 + threadIdx.x;
    if (r >= 841) return;
    float x0 = biases[r * 2], x1 = biases[r * 2 + 1];
    float t[24], u[24];
    #pragma unroll
    for (int i = 0; i < 24; ++i) t[i] = ppw[i * 2] * x0 + ppw[i * 2 + 1] * x1 + ppb[i];

    ln24_relu(t, g1, b1);
    #pragma unroll
    for (int i = 0; i < 24; ++i) {
        float s = bb1[i];
        for (int j = 0; j < 24; ++j) s += w1[i * 24 + j] * t[j];
        u[i] = s;
    }
    ln24_relu(u, g2, b2);
    #pragma unroll
    for (int i = 0; i < 24; ++i) {
        float s = bb2[i];
        for (int j = 0; j < 24; ++j) s += w2[i * 24 + j] * u[j];
        t[i] = s;
    }
    ln24_relu(t, g3, b3);
    #pragma unroll
    for (int o = 0; o < 8; ++o) {
        float s = bb3[o];
        for (int j = 0; j < 24; ++j) s += w3[o * 24 + j] * t[j];
        pos[r * 8 + o] = s;
    }
}

// ---------------------------------------------------------------------------
// Split qkv[12608][1152] into head-major padded Qn/Kn/V [B][H][208][48],
// L2-normalizing q and k rows; pad rows (n>=197) are zeroed so attention
// tiles are always full (WMMA needs EXEC all-1s / no tail logic).
// ---------------------------------------------------------------------------
__global__ __launch_bounds__(256) void qkv_split_norm(
    const float* __restrict__ qkv,
    float* __restrict__ Qn, float* __restrict__ Kn, float* __restrict__ Vv)
{
    int gid = blockIdx.x * 256 + threadIdx.x;
    int lane16 = gid & 15;
    int rowId = gid >> 4;
    if (rowId >= BSZ * NH * NPAD) return;
    int n = rowId % NPAD;
    int h = (rowId / NPAD) % NH;
    int b = rowId / (NPAD * NH);
    size_t ob = (size_t)rowId * HD;
    if (n >= NTOK) {
        #pragma unroll
        for (int i = 0; i < 3; ++i) {
            Qn[ob + lane16 + i * 16] = 0.f;
            Kn[ob + lane16 + i * 16] = 0.f;
            Vv[ob + lane16 + i * 16] = 0.f;
        }
        return;
    }
    size_t ib = ((size_t)(b * NTOK + n)) * (3 * CDIM) + h * HD;
    float q[3], k[3], sq = 0.f, sk = 0.f;
    #pragma unroll
    for (int i = 0; i < 3; ++i) {
        q[i] = qkv[ib + lane16 + i * 16];
        k[i] = qkv[ib + CDIM + lane16 + i * 16];
        sq += q[i] * q[i]; sk += k[i] * k[i];
    }
    #pragma unroll
    for (int off = 1; off < 16; off <<= 1) {  // stays inside the 16-lane group
        sq += __shfl_xor(sq, off, 32);
        sk += __shfl_xor(sk, off, 32);
    }
    float invq = 1.f / fmaxf(sqrtf(sq), 1e-12f);
    float invk = 1.f / fmaxf(sqrtf(sk), 1e-12f);
    #pragma unroll
    for (int i = 0; i < 3; ++i) {
        Qn[ob + lane16 + i * 16] = q[i] * invq;
        Kn[ob + lane16 + i * 16] = k[i] * invk;
        Vv[ob + lane16 + i * 16] = qkv[ib + 2 * CDIM + lane16 + i * 16];
    }
}

// ---------------------------------------------------------------------------
// Fused attention, one block per (16-row tile, batch). 8 waves = 8 heads.
// LDS: scores[8][16][208] + probs[8][16][208] = 213 KB (CDNA5: 320 KB/WG).
// Phase 1 (wave=h):  S_h = (Qn Kn^T)*scale_h + rpb_h              -> scores
// Phase 2 (wave=o):  premix over h, adjacency mask, softmax       -> probs
// Phase 2b (wave=o): post-softmax head mix P'_o = sum_h post[o,h]P_h -> scores
// Phase 3 (wave=o):  ctx_o = P'_o @ V_o via WMMA.
// ---------------------------------------------------------------------------
__global__ __launch_bounds__(256) void attn_fused(
    const float* __restrict__ Qn, const float* __restrict__ Kn,
    const float* __restrict__ Vv, const float* __restrict__ pos,
    const int* __restrict__ rel_index, const float* __restrict__ adj,
    const float* __restrict__ scale, const float* __restrict__ pre_w,
    const float* __restrict__ post_w, float* __restrict__ ctx)
{
    extern __shared__ float lds[];
    float* scores = lds;                    // [8][16][NPAD]
    float* probs  = lds + NH * 16 * NPAD;   // [8][16][NPAD]

    const int tid  = threadIdx.x;
    const int wave = tid >> 5;
    const int lane = tid & 31;
    const int lm   = lane & 15;
    const int hi   = lane >> 4;
    const int koff = hi * 2;
    const int tr   = blockIdx.x;   // 0..12
    const int b    = blockIdx.y;   // 0..63

    // ---- Phase 1: per-head raw scores via WMMA ----
    {
        const int h = wave;
        const float* qb = Qn + (((size_t)b * NH + h) * NPAD) * HD;
        const float* kb = Kn + (((size_t)b * NH + h) * NPAD) * HD;
        const float sc = fminf(scale[h], 100.f);
        const int rowq = tr * 16 + lm;
        v2f aq[12];
        #pragma unroll
        for (int kk = 0; kk < 12; ++kk) {
            aq[kk].x = qb[rowq * HD + kk * 4 + koff];
            aq[kk].y = qb[rowq * HD + kk * 4 + koff + 1];
        }
        for (int t = 0; t < 13; ++t) {
            v8f d = {};
            #pragma unroll
            for (int kk = 0; kk < 12; ++kk) {
                v2f bf;
                bf.x = kb[(t * 16 + lm) * HD + kk * 4 + koff];
                bf.y = kb[(t * 16 + lm) * HD + kk * 4 + koff + 1];
                d = __builtin_amdgcn_wmma_f32_16x16x4_f32(
                    false, aq[kk], false, bf, (short)0, d, false, false);
            }
            #pragma unroll
            for (int i = 0; i < 8; ++i) {
                int rl = i + (hi << 3);
                int n  = tr * 16 + rl;
                int m  = t * 16 + lm;
                int nn = (n < NTOK) ? n : 0;
                int mm = (m < NTOK) ? m : 0;
                float rpb = pos[rel_index[nn * NTOK + mm] * NH + h];
                float v = d[i] * sc + ((n < NTOK && m < NTOK) ? rpb : 0.f);
                scores[(h * 16 + rl) * NPAD + m] = v;
            }
        }
    }
    __syncthreads();

    // ---- Phase 2: premix + mask + softmax (per output head o) ----
    {
        const int o = wave;
        float pw[8];
        #pragma unroll
        for (int h = 0; h < 8; ++h) pw[h] = pre_w[o * 8 + h];
        const int row = lm;
        const int n = tr * 16 + row;
        const float* adjrow = adj + ((size_t)o * NTOK + ((n < NTOK) ? n : 0)) * NTOK;
        float maxv = -3.0e38f;
        for (int m = hi; m < NPAD; m += 2) {
            float s = 0.f;
            #pragma unroll
            for (int h = 0; h < 8; ++h) s += pw[h] * scores[(h * 16 + row) * NPAD + m];
            int mm = (m < NTOK) ? m : 0;
            float am = adjrow[mm];
            bool valid = (n < NTOK) && (m < NTOK) && (am > 0.f);
            s = valid ? s : -1.0e9f;
            probs[(o * 16 + row) * NPAD + m] = s;
            maxv = fmaxf(maxv, s);
        }
        maxv = fmaxf(maxv, __shfl_xor(maxv, 16, 32));
        float sum = 0.f;
        for (int m = hi; m < NPAD; m += 2) {
            float e = __expf(probs[(o * 16 + row) * NPAD + m] - maxv);
            probs[(o * 16 + row) * NPAD + m] = e;
            sum += e;
        }
        sum += __shfl_xor(sum, 16, 32);
        float inv = 1.f / sum;
        for (int m = hi; m < NPAD; m += 2)
            probs[(o * 16 + row) * NPAD + m] *= inv;
    }
    __syncthreads();

    // ---- Phase 2b: post-softmax head mixing, written into scores buffer ----
    {
        const int o = wave;
        float pw[8];
        #pragma unroll
        for (int h = 0; h < 8; ++h) pw[h] = post_w[o * 8 + h];
        const int row = lm;
        for (int m = hi; m < NPAD; m += 2) {
            float s = 0.f;
            #pragma unroll
            for (int h = 0; h < 8; ++h) s += pw[h] * probs[(h * 16 + row) * NPAD + m];
            scores[(o * 16 + row) * NPAD + m] = s;
        }
    }
    __syncthreads();

    // ---- Phase 3: ctx_o = P'_o @ V_o via WMMA ----
    {
        const int o = wave;
        const float* vb = Vv + (((size_t)b * NH + o) * NPAD) * HD;
        v8f acc[3] = {};
        for (int k4 = 0; k4 < NPAD; k4 += 4) {
            v2f a;
            a.x = scores[(o * 16 + lm) * NPAD + k4 + koff];
            a.y = scores[(o * 16 + lm) * NPAD + k4 + koff + 1];
            #pragma unroll
            for (int j = 0; j < 3; ++j) {
                v2f bf;
                bf.x = vb[(k4 + koff) * HD + j * 16 + lm];
                bf.y = vb[(k4 + koff + 1) * HD + j * 16 + lm];
                acc[j] = __builtin_amdgcn_wmma_f32_16x16x4_f32(
                    false, a, false, bf, (short)0, acc[j], false, false);
            }
        }
        #pragma unroll
        for (int j = 0; j < 3; ++j) {
            #pragma unroll
            for (int i = 0; i < 8; ++i) {
                int n = tr * 16 + i + (hi << 3);
                if (n < NTOK)
                    ctx[((size_t)b * NTOK + n) * CDIM + o * HD + j * 16 + lm] = acc[j][i];
            }
        }
    }
}

// ---------------------------------------------------------------------------
extern "C" void kernel_launch(void* const* d_in, const int* in_sizes, int n_in,
                              void* d_out, int out_size, void* d_ws, size_t ws_size,
                              hipStream_t stream)
{
    (void)in_sizes; (void)n_in; (void)out_size; (void)ws_size;
    const float* x      = (const float*)d_in[0];
    const float* qkv_w  = (const float*)d_in[1];
    const float* scale  = (const float*)d_in[2];
    const float* ppw    = (const float*)d_in[3];
    const float* ppb    = (const float*)d_in[4];
    const float* g1  = (const float*)d_in[5];
    const float* b1  = (const float*)d_in[6];
    const float* w1  = (const float*)d_in[7];
    const float* bb1 = (const float*)d_in[8];
    const float* g2  = (const float*)d_in[9];
    const float* b2  = (const float*)d_in[10];
    const float* w2  = (const float*)d_in[11];
    const float* bb2 = (const float*)d_in[12];
    const float* g3  = (const float*)d_in[13];
    const float* b3  = (const float*)d_in[14];
    const float* w3  = (const float*)d_in[15];
    const float* bb3 = (const float*)d_in[16];
    const float* pre_w  = (const float*)d_in[17];
    const float* post_w = (const float*)d_in[18];
    const float* proj_w = (const float*)d_in[19];
    const float* proj_b = (const float*)d_in[20];
    const float* biases = (const float*)d_in[21];
    const float* adj    = (const float*)d_in[22];
    const int*   rel_index = (const int*)d_in[23];

    float* ws  = (float*)d_ws;
    float* pos = ws;                                       //  841*8
    float* qkv = ws + 8192;                                // 12608*1152
    float* Qn  = qkv + (size_t)MROWS * 3 * CDIM;           // 64*8*208*48
    float* Kn  = Qn + (size_t)BSZ * NH * NPAD * HD;
    float* Vv  = Kn + (size_t)BSZ * NH * NPAD * HD;
    float* ctx = Vv + (size_t)BSZ * NH * NPAD * HD;        // 12608*384

    pos_mlp<<<7, 128, 0, stream>>>(biases, ppw, ppb, g1, b1, w1, bb1,
                                   g2, b2, w2, bb2, g3, b3, w3, bb3, pos);

    gemm_wmma_f32<<<dim3((3 * CDIM) / 128, (MROWS + 127) / 128), 256, 0, stream>>>(
        x, qkv_w, nullptr, qkv, MROWS, 3 * CDIM, CDIM);

    qkv_split_norm<<<(BSZ * NH * NPAD * 16) / 256, 256, 0, stream>>>(qkv, Qn, Kn, Vv);

    size_t ldsBytes = (size_t)2 * NH * 16 * NPAD * sizeof(float);  // 213 KB (< 320 KB/WG on CDNA5)
    attn_fused<<<dim3(13, BSZ), 256, ldsBytes, stream>>>(
        Qn, Kn, Vv, pos, rel_index, adj, scale, pre_w, post_w, ctx);

    gemm_wmma_f32<<<dim3(CDIM / 128, (MROWS + 127) / 128), 256, 0, stream>>>(
        ctx, proj_w, proj_b, (float*)d_out, MROWS, CDIM, CDIM);
}